// FFTlayer_52699248722191
// MI455X (gfx1250) — compile-verified
//
#include <hip/hip_runtime.h>
#include <math.h>

// ---------------------------------------------------------------------------
// FFT(4096) along axis 1 of x[64][4096][128] -> (re, im) f32 planes.
//
// Roofline (MI455X): 134 MB in + 268 MB out ~= 402 MB -> ~17 us floor at
// 23.3 TB/s; FLOPs (~13 GF in radix-16 matmul form) are negligible. Single
// fused pass: each workgroup owns a (batch b, 8-wide m-tile) slab, stages all
// 4096 points x 8 lines in LDS, runs three radix-16 stages as complex 16x16
// matmuls on the f32 matrix pipe (V_WMMA_F32_16X16X4_F32 chained x4 over K),
// and streams results to HBM with non-temporal stores.
//
// CDNA5 data movement: the input slab (4096 rows x 8 contiguous floats,
// row stride 128 floats) is a textbook 2D TDM tile, so wave 0 issues a single
// tensor_load_to_lds (Tensor Data Mover DMA, tracked by TENSORcnt) and the
// other waves build the twiddle table while the DMA is in flight.
// This toolchain exposes the 6-arg builtin:
//   (uint32x4 g0, int32x8 g1, int32x4 g2, int32x4 g3, int32x8, i32 cpol)
//
// LDS budget (CDNA5 WGP has 320 KB):
//   data re plane  128 KB | data im plane  128 KB | twiddle table 32 KB = 288 KB
//
// All twiddles are integer powers of W_4096 = exp(-2*pi*i/4096); trig runs
// once per workgroup into the table (exactly-reduced small args -> hw
// v_sin/v_cos), stages and DFT16 fragments are pure table lookups.
//
// Math (DIF, N = 16*16*16):
//   stage1: y1[a,s]    = sum_r D16[a,r] * x[r*256+s]        then * W4096^(a*s)
//   stage2: y2[a,b,s2] = sum_r D16[b,r] * y1[a, r*16+s2]    then * W256^(b*s2)
//   stage3: y3[a,b,c]  = sum_r D16[c,r] * y2[a,b,r]
//   X[a + 16*b + 256*c] = y3[a,b,c]                  (digit-reversed store)
// Stage 1 exploits the purely-real input (half the WMMA work).
// ---------------------------------------------------------------------------

#define NFFT    4096
#define MTOT    128
#define MT      8        // m-columns per workgroup
#define LROW    8        // floats per n-row in LDS (== MT)
#define THREADS 256      // 8 wave32 waves
#define NWAVE   8

typedef __attribute__((ext_vector_type(2))) float v2f;
typedef __attribute__((ext_vector_type(4))) float v4f;
typedef __attribute__((ext_vector_type(8))) float v8f;
typedef __attribute__((ext_vector_type(4))) unsigned int v4u;
typedef __attribute__((ext_vector_type(4))) int v4i;
typedef __attribute__((ext_vector_type(8))) int v8i;

static __device__ __forceinline__ v8f wmma4(v2f a, v2f b, v8f c) {
  // D(16x16,f32) += A(16x4,f32) * B(4x16,f32); full K=16 via 4 chained calls.
  // NEG must be 0 for f32 per ISA; negation handled via pre-negated fragments.
  return __builtin_amdgcn_wmma_f32_16x16x4_f32(false, a, false, b, (short)0, c,
                                               false, false);
}

template <int STAGE>
static __device__ __forceinline__ void radix16_stage(
    float* __restrict__ Lre, float* __restrict__ Lim,
    const float* __restrict__ Ltw,
    const v2f* Ar, const v2f* Ai, const v2f* Ani,
    int wave, int lane,
    float* __restrict__ out_re, float* __restrict__ out_im,
    size_t obase)
{
  const int stride = (STAGE == 1) ? 256 : (STAGE == 2) ? 16 : 1;
  const int twsc   = (STAGE == 1) ? 1 : 16;   // table index scale (4096/Ntw)
  const int hi  = lane >> 4;      // wave32: upper half-lanes hold rows+8 / K+2
  const int col = lane & 15;

  for (int t = wave; t < 128; t += NWAVE) {
    const int g  = t * 16 + col;          // global column id
    const int mi = g & 7;                 // m within tile
    const int h  = g >> 3;                // batch-digit part of the column
    int base, tws;
    if (STAGE == 1)      { base = h;                              tws = h;      }
    else if (STAGE == 2) { base = (h >> 4) * 256 + (h & 15);      tws = h & 15; }
    else                 { base = (h >> 4) * 256 + (h & 15) * 16; tws = 0;      }

    // B fragments: logical tile rows r = FFT input digit, 16 cols.
    // For K-chunk kc this lane supplies rows r0 = 4*kc + 2*hi and r0+1.
    v2f Bxr[4], Bxi[4];
#pragma unroll
    for (int kc = 0; kc < 4; ++kc) {
      const int n0 = base + (kc * 4 + 2 * hi) * stride;
      const int n1 = n0 + stride;
      v2f r; r.x = Lre[n0 * LROW + mi]; r.y = Lre[n1 * LROW + mi]; Bxr[kc] = r;
      if (STAGE != 1) {
        v2f q; q.x = Lim[n0 * LROW + mi]; q.y = Lim[n1 * LROW + mi]; Bxi[kc] = q;
      }
    }

    // Hoist twiddle table loads: indices are accumulator-independent, so these
    // LDS loads can overlap the WMMA chain instead of serializing the epilogue.
    float twc[8], twn[8];
    if (STAGE != 3) {
#pragma unroll
      for (int v = 0; v < 8; ++v) {
        const int j = v + 8 * hi;
        const int p = (j * tws * twsc) & (NFFT - 1);   // W_4096^p
        twc[v] = Ltw[2 * p];
        twn[v] = Ltw[2 * p + 1];
      }
    }

    // Complex matmul: Yr = Dr*Xr - Di*Xi ; Yi = Dr*Xi + Di*Xr
    v8f yr = {}, yi = {};
#pragma unroll
    for (int kc = 0; kc < 4; ++kc) yr = wmma4(Ar[kc], Bxr[kc], yr);
    if (STAGE != 1) {
#pragma unroll
      for (int kc = 0; kc < 4; ++kc) yr = wmma4(Ani[kc], Bxi[kc], yr);
#pragma unroll
      for (int kc = 0; kc < 4; ++kc) yi = wmma4(Ar[kc], Bxi[kc], yi);
    }
#pragma unroll
    for (int kc = 0; kc < 4; ++kc) yi = wmma4(Ai[kc], Bxr[kc], yi);

    // Twiddle + store. C/D layout: VGPR v holds row j = v + 8*hi, col = lane&15.
#pragma unroll
    for (int v = 0; v < 8; ++v) {
      const int j = v + 8 * hi;
      const float zr = yr[v], zi = yi[v];
      if (STAGE != 3) {
        const float tr = zr * twc[v] - zi * twn[v];
        const float ti = zr * twn[v] + zi * twc[v];
        const int n_out = base + j * stride;  // in-place (tile owns its cols)
        Lre[n_out * LROW + mi] = tr;
        Lim[n_out * LROW + mi] = ti;
      } else {
        const int a  = h >> 4;
        const int bd = h & 15;
        const int k  = a + 16 * bd + 256 * j; // digit-reversed frequency index
        __builtin_nontemporal_store(zr, &out_re[obase + (size_t)k * MTOT + mi]);
        __builtin_nontemporal_store(zi, &out_im[obase + (size_t)k * MTOT + mi]);
      }
    }
  }
}

__global__ void __launch_bounds__(THREADS)
fft4096_wmma_kernel(const float* __restrict__ x,
                    float* __restrict__ out_re,
                    float* __restrict__ out_im)
{
  extern __shared__ float lds[];           // 288 KB total
  float* Lre = lds;                        // 4096*8 floats (TDM dest, offset 0)
  float* Lim = lds + NFFT * LROW;          // 4096*8 floats
  float* Ltw = lds + 2 * NFFT * LROW;      // 4096 x (cos, sin)

  const int b    = blockIdx.y;
  const int m0   = blockIdx.x * MT;
  const int tid  = threadIdx.x;
  const int wave = tid >> 5;               // gfx1250 is wave32
  const int lane = tid & 31;

  const size_t ibase = ((size_t)b * NFFT) * MTOT + m0;

  // ---- TDM: one DMA moves the whole (4096 x 8 f32, row stride 128) tile ----
  // Issued by wave 0 only (TDM ignores EXEC; wave-granular branch prevents 8
  // duplicate DMAs). Other waves overlap twiddle-table building with the DMA.
  if (wave == 0) {
    const unsigned long long ga = (unsigned long long)(const void*)(x + ibase);
    v4u g0;
    g0[0] = 1u;                              // count=1, user descriptor
    g0[1] = 0u;                              // lds_addr = 0 (Lre at base)
    g0[2] = (unsigned)(ga & 0xffffffffu);    // global_addr[31:0]
    g0[3] = (unsigned)((ga >> 32) & 0x01ffffffu) | 0x80000000u;  // [56:32]|type=2
    v8i g1;
    g1[0] = 0x20000;                         // data_size=2 (4 B), no multicast
    g1[1] = (int)(8u << 16);                 // tensor_dim0 = 8
    g1[2] = (int)((unsigned)NFFT << 16);     // tensor_dim1 = 4096 (low half)
    g1[3] = (int)(8u << 16);                 // tile_dim0 = 8
    g1[4] = NFFT;                            // tile_dim1 = 4096, tile_dim2 = 0
    g1[5] = MTOT;                            // tensor_dim0_stride = 128
    g1[6] = 0;                               // stride hi / dim1_stride lo
    g1[7] = 0;
    const v4i gz4 = {0, 0, 0, 0};            // 2D tile: groups 2/3 unused
    const v8i gz8 = {0, 0, 0, 0, 0, 0, 0, 0};
    __builtin_amdgcn_tensor_load_to_lds(g0, g1, gz4, gz4, gz8, 0);
  }

  // ---- build twiddle table: Ltw[2p..2p+1] = exp(-i*2*pi*p/4096) ----
  // Small exactly-reduced args -> fast hw v_sin/v_cos path; overlaps the DMA.
  for (int p = tid; p < NFFT; p += THREADS) {
    float sn, cs;
    __sincosf(-1.5339807878856412e-3f * (float)p, &sn, &cs);  // -2*pi/4096 * p
    Ltw[2 * p]     = cs;
    Ltw[2 * p + 1] = sn;
  }

  if (wave == 0) {
    __builtin_amdgcn_s_wait_tensorcnt(0);    // DMA complete before barrier
  }
  // Lim needs no init: stage 1 treats the input as purely real and fully
  // overwrites both planes.

  __syncthreads();

  // ---- per-lane DFT16 A-matrix fragments, read from the twiddle table ----
  // A(16x4,f32) layout: lanes 0-15 -> row L, K = {4kc, 4kc+1};
  //                     lanes 16-31 -> row L-16, K = {4kc+2, 4kc+3}.
  // D16[a,r] = exp(-i*2*pi*(a*r mod 16)/16) = W_4096^(256*(a*r mod 16)).
  const int arow = lane & 15;
  const int hi   = lane >> 4;
  v2f Ar[4], Ai[4], Ani[4];
#pragma unroll
  for (int kc = 0; kc < 4; ++kc) {
    const int c0 = kc * 4 + 2 * hi;
    const int p0 = ((arow * c0) & 15) * 256;
    const int p1 = ((arow * (c0 + 1)) & 15) * 256;
    const float cs0 = Ltw[2 * p0], sn0 = Ltw[2 * p0 + 1];
    const float cs1 = Ltw[2 * p1], sn1 = Ltw[2 * p1 + 1];
    v2f r;  r.x  = cs0;  r.y  = cs1;  Ar[kc]  = r;   //  Re(D16)
    v2f q;  q.x  = sn0;  q.y  = sn1;  Ai[kc]  = q;   //  Im(D16) (= -sin)
    v2f nn; nn.x = -sn0; nn.y = -sn1; Ani[kc] = nn;  // -Im(D16)
  }

  const size_t obase = ((size_t)b * NFFT) * MTOT + m0;

  radix16_stage<1>(Lre, Lim, Ltw, Ar, Ai, Ani, wave, lane, out_re, out_im, obase);
  __syncthreads();
  radix16_stage<2>(Lre, Lim, Ltw, Ar, Ai, Ani, wave, lane, out_re, out_im, obase);
  __syncthreads();
  radix16_stage<3>(Lre, Lim, Ltw, Ar, Ai, Ani, wave, lane, out_re, out_im, obase);
}

extern "C" void kernel_launch(void* const* d_in, const int* in_sizes, int n_in,
                              void* d_out, int out_size, void* d_ws, size_t ws_size,
                              hipStream_t stream) {
  (void)in_sizes; (void)n_in; (void)out_size; (void)d_ws; (void)ws_size;
  const float* x = (const float*)d_in[0];
  float* out    = (float*)d_out;
  float* out_re = out;                                  // real plane first
  float* out_im = out + (size_t)64 * NFFT * MTOT;       // then imag plane

  dim3 grid(MTOT / MT, 64, 1);                          // 16 m-tiles x 64 batches
  const size_t shmem =
      (size_t)(2 * NFFT * LROW + 2 * NFFT) * sizeof(float);  // 288 KB
  fft4096_wmma_kernel<<<grid, dim3(THREADS, 1, 1), shmem, stream>>>(
      x, out_re, out_im);
}